// test_reproject_90202903150676
// MI455X (gfx1250) — compile-verified
//
#include <hip/hip_runtime.h>
#include <math.h>

typedef __attribute__((ext_vector_type(2))) float v2f;
typedef __attribute__((ext_vector_type(4))) float v4f;
typedef __attribute__((ext_vector_type(8))) float v8f;

#define BB 2
#define CC 32
#define DD 32
#define HH 128
#define WW 160
#define HW (HH * WW)

// ---------------------------------------------------------------------------
// Setup kernel: one wave32. Computes per-batch rot (3x3) and trans (3) via
// four V_WMMA_F32_16X16X4_F32 issues, both batches packed into one tile:
//   batch0 in rows/cols 0..3, batch1 in rows 8..10 / cols 8..11
// (so both batches' results land in D VGPRs 0..2; cross blocks are ignored).
// Chain:  P  = Kinv * [srcP(3x4) | refP(3x4)]          (WMMA 1)
//         proj = srcR * refR^T                          (WMMA 2)
//         Kp = K * proj ,  Kst = K * src_t              (WMMA 3)
//         rot = Kp * Kinv , Kpt = Kp * ref_t            (WMMA 4)
//         trans = Kst - Kpt
// ws layout: [b][0..8]=rot row-major, [b][9..11]=trans
// ---------------------------------------------------------------------------
__global__ __launch_bounds__(32) void setup_rt_kernel(
    const float* __restrict__ intr,      // (B,3,3)
    const float* __restrict__ intr_inv,  // (B,3,3)
    const float* __restrict__ srcp,      // (B,4,4)
    const float* __restrict__ refp,      // (B,4,4)
    float* __restrict__ ws)              // 24 floats
{
    __shared__ float sK[18], sKi[18], sSP[32], sRP[32];
    __shared__ float sSp34[24], sRp34[24];   // src_p, ref_p : [b][3][4]
    __shared__ float sProj[18];              // [b][3][3]
    __shared__ float sKp[18];                // [b][3][3]
    __shared__ float sKst[6];                // [b][3]

    const int tid  = threadIdx.x;
    const int rowA = tid & 15;   // A-matrix row / B,D column (N)
    const int kh   = tid >> 4;   // K-half: VGPR0/1 hold K = 2*kh, 2*kh+1

    if (tid < 18) { sK[tid] = intr[tid]; sKi[tid] = intr_inv[tid]; }
    sSP[tid] = srcp[tid];
    sRP[tid] = refp[tid];
    __syncthreads();

    v8f czero = {};

    // ---- WMMA 1: D1 = Kinv_blk(16x4) * [srcP|refP]_blk(4x16) --------------
    auto A1 = [&](int r, int k) -> float {
        if (k >= 3) return 0.f;
        if (r < 3)            return sKi[r * 3 + k];
        if (r >= 8 && r < 11) return sKi[9 + (r - 8) * 3 + k];
        return 0.f;
    };
    auto B1 = [&](int k, int n) -> float {
        if (k >= 3) return 0.f;
        int g = n >> 2, j = n & 3;
        if (g == 0) return sSP[k * 4 + j];          // srcP b0 rows 0..2
        if (g == 1) return sRP[k * 4 + j];          // refP b0
        if (g == 2) return sSP[16 + k * 4 + j];     // srcP b1
        return sRP[16 + k * 4 + j];                 // refP b1
    };
    v2f a, b;
    a[0] = A1(rowA, 2 * kh); a[1] = A1(rowA, 2 * kh + 1);
    b[0] = B1(2 * kh, rowA); b[1] = B1(2 * kh + 1, rowA);
    v8f d = __builtin_amdgcn_wmma_f32_16x16x4_f32(false, a, false, b,
                                                  (short)0, czero, false, false);
    {
        int n = rowA;
        #pragma unroll
        for (int r = 0; r < 8; ++r) {
            int m = r + 8 * kh;
            float v = d[r];
            if (kh == 0 && m < 3) {
                if (n < 4)      sSp34[m * 4 + n] = v;
                else if (n < 8) sRp34[m * 4 + (n - 4)] = v;
            } else if (kh == 1 && m >= 8 && m < 11) {
                if (n >= 8 && n < 12)  sSp34[12 + (m - 8) * 4 + (n - 8)]  = v;
                else if (n >= 12)      sRp34[12 + (m - 8) * 4 + (n - 12)] = v;
            }
        }
    }
    __syncthreads();

    // ---- WMMA 2: proj = srcR * refR^T -------------------------------------
    auto A2 = [&](int r, int k) -> float {
        if (k >= 3) return 0.f;
        if (r < 3)            return sSp34[r * 4 + k];
        if (r >= 8 && r < 11) return sSp34[12 + (r - 8) * 4 + k];
        return 0.f;
    };
    auto B2 = [&](int k, int n) -> float {   // refR^T: B(k,n) = refR[n][k]
        if (k >= 3) return 0.f;
        if (n < 3)            return sRp34[n * 4 + k];
        if (n >= 8 && n < 11) return sRp34[12 + (n - 8) * 4 + k];
        return 0.f;
    };
    a[0] = A2(rowA, 2 * kh); a[1] = A2(rowA, 2 * kh + 1);
    b[0] = B2(2 * kh, rowA); b[1] = B2(2 * kh + 1, rowA);
    d = __builtin_amdgcn_wmma_f32_16x16x4_f32(false, a, false, b,
                                              (short)0, czero, false, false);
    {
        int n = rowA;
        #pragma unroll
        for (int r = 0; r < 8; ++r) {
            int m = r + 8 * kh;
            float v = d[r];
            if (kh == 0 && m < 3 && n < 3)                       sProj[m * 3 + n] = v;
            else if (kh == 1 && m >= 8 && m < 11 && n >= 8 && n < 11)
                sProj[9 + (m - 8) * 3 + (n - 8)] = v;
        }
    }
    __syncthreads();

    // ---- WMMA 3: Kp = K * proj ; Kst = K * src_t --------------------------
    auto A3 = [&](int r, int k) -> float {
        if (k >= 3) return 0.f;
        if (r < 3)            return sK[r * 3 + k];
        if (r >= 8 && r < 11) return sK[9 + (r - 8) * 3 + k];
        return 0.f;
    };
    auto B3 = [&](int k, int n) -> float {
        if (k >= 3) return 0.f;
        if (n < 3)             return sProj[k * 3 + n];
        if (n == 3)            return sSp34[k * 4 + 3];            // src_t b0
        if (n >= 8 && n < 11)  return sProj[9 + k * 3 + (n - 8)];
        if (n == 11)           return sSp34[12 + k * 4 + 3];       // src_t b1
        return 0.f;
    };
    a[0] = A3(rowA, 2 * kh); a[1] = A3(rowA, 2 * kh + 1);
    b[0] = B3(2 * kh, rowA); b[1] = B3(2 * kh + 1, rowA);
    d = __builtin_amdgcn_wmma_f32_16x16x4_f32(false, a, false, b,
                                              (short)0, czero, false, false);
    {
        int n = rowA;
        #pragma unroll
        for (int r = 0; r < 8; ++r) {
            int m = r + 8 * kh;
            float v = d[r];
            if (kh == 0 && m < 3) {
                if (n < 3)       sKp[m * 3 + n] = v;
                else if (n == 3) sKst[m] = v;
            } else if (kh == 1 && m >= 8 && m < 11) {
                if (n >= 8 && n < 11) sKp[9 + (m - 8) * 3 + (n - 8)] = v;
                else if (n == 11)     sKst[3 + (m - 8)] = v;
            }
        }
    }
    __syncthreads();

    // ---- WMMA 4: rot = Kp * Kinv ; Kpt = Kp * ref_t -----------------------
    auto A4 = [&](int r, int k) -> float {
        if (k >= 3) return 0.f;
        if (r < 3)            return sKp[r * 3 + k];
        if (r >= 8 && r < 11) return sKp[9 + (r - 8) * 3 + k];
        return 0.f;
    };
    auto B4 = [&](int k, int n) -> float {
        if (k >= 3) return 0.f;
        if (n < 3)             return sKi[k * 3 + n];
        if (n == 3)            return sRp34[k * 4 + 3];            // ref_t b0
        if (n >= 8 && n < 11)  return sKi[9 + k * 3 + (n - 8)];
        if (n == 11)           return sRp34[12 + k * 4 + 3];       // ref_t b1
        return 0.f;
    };
    a[0] = A4(rowA, 2 * kh); a[1] = A4(rowA, 2 * kh + 1);
    b[0] = B4(2 * kh, rowA); b[1] = B4(2 * kh + 1, rowA);
    d = __builtin_amdgcn_wmma_f32_16x16x4_f32(false, a, false, b,
                                              (short)0, czero, false, false);
    {
        int n = rowA;
        #pragma unroll
        for (int r = 0; r < 8; ++r) {
            int m = r + 8 * kh;
            float v = d[r];
            if (kh == 0 && m < 3) {
                if (n < 3)       ws[m * 3 + n] = v;                 // rot b0
                else if (n == 3) ws[9 + m] = sKst[m] - v;           // trans b0
            } else if (kh == 1 && m >= 8 && m < 11) {
                if (n >= 8 && n < 11) ws[12 + (m - 8) * 3 + (n - 8)] = v;
                else if (n == 11)     ws[12 + 9 + (m - 8)] = sKst[3 + (m - 8)] - v;
            }
        }
    }
}

// ---------------------------------------------------------------------------
// Bulk kernel: HBM-store-bound homography warp + bilinear gather.
// grid.y = B*D slice, grid.x * 256 threads * 4 pixels covers H*W.
// Projection + weights computed once per pixel, reused across 32 channels.
// Output stored non-temporally (write-once, keep L2 for the gather sources).
// ---------------------------------------------------------------------------
__global__ __launch_bounds__(256) void warp_bilinear_kernel(
    const float* __restrict__ feat,   // (B,C,H,W)
    const float* __restrict__ depth,  // (B,D,H,W)
    const float* __restrict__ rt,     // ws: [b][12]
    float* __restrict__ out)          // (B,C,D,H,W)
{
    const int bd = blockIdx.y;
    const int bidx = bd >> 5;           // / DD
    const int dsl  = bd & 31;           // % DD
    const int pix0 = (blockIdx.x * 256 + threadIdx.x) * 4;

    const float* __restrict__ R = rt + bidx * 12;
    const float r00 = R[0], r01 = R[1], r02 = R[2];
    const float r10 = R[3], r11 = R[4], r12 = R[5];
    const float r20 = R[6], r21 = R[7], r22 = R[8];
    const float t0 = R[9], t1 = R[10], t2 = R[11];

    const v4f dv = *(const v4f*)(depth + ((size_t)(bidx * DD + dsl)) * HW + pix0);

    const int hy  = pix0 / WW;          // 4 consecutive pixels share the row
    const int wx0 = pix0 - hy * WW;
    const float y = (float)hy;

    float wa[4], wb[4], wc[4], wd[4];
    int   i00[4], i10[4], i01[4], i11[4];

    #pragma unroll
    for (int p = 0; p < 4; ++p) {
        const float x  = (float)(wx0 + p);
        const float rx = r00 * x + r01 * y + r02;
        const float ry = r10 * x + r11 * y + r12;
        const float rz = r20 * x + r21 * y + r22;
        const float de = dv[p];
        float X = rx * de + t0;
        float Y = ry * de + t1;
        float Z = rz * de + t2;
        const float m = (Z > 0.001f) ? 1.f : 0.f;
        X *= m; Y *= m; Z = Z * m + (1.f - m);
        float px = X / Z;
        float py = Y / Z;
        px = px * (float)(px < (float)WW) * (float)(px >= 0.f);
        py = py * (float)(py < (float)HH) * (float)(py >= 0.f);
        const float x0 = floorf(px), y0 = floorf(py);
        const float x1 = x0 + 1.f,   y1 = y0 + 1.f;
        wa[p] = (x1 - px) * (y1 - py);
        wb[p] = (x1 - px) * (py - y0);
        wc[p] = (px - x0) * (y1 - py);
        wd[p] = (px - x0) * (py - y0);
        const int x0i = (int)x0;
        const int y0i = (int)y0;
        const int x1i = (int)(x1 - floorf(x1 / (float)WW));  // W wrap (x1==W -> W-1)
        const int y1i = (int)(y1 - floorf(y1 / (float)HH));
        i00[p] = y0i * WW + x0i;
        i10[p] = y1i * WW + x0i;
        i01[p] = y0i * WW + x1i;
        i11[p] = y1i * WW + x1i;
    }

    const float* __restrict__ fb = feat + (size_t)bidx * CC * HW;
    float* __restrict__ ob = out + (((size_t)bidx * CC) * DD + dsl) * HW + pix0;

    #pragma unroll 4
    for (int ch = 0; ch < CC; ++ch) {
        const float* __restrict__ f = fb + (size_t)ch * HW;
        v4f res;
        #pragma unroll
        for (int p = 0; p < 4; ++p) {
            res[p] = f[i00[p]] * wa[p] + f[i10[p]] * wb[p]
                   + f[i01[p]] * wc[p] + f[i11[p]] * wd[p];
        }
        __builtin_nontemporal_store(res, (v4f*)(ob + (size_t)ch * DD * HW));
    }
}

extern "C" void kernel_launch(void* const* d_in, const int* in_sizes, int n_in,
                              void* d_out, int out_size, void* d_ws, size_t ws_size,
                              hipStream_t stream) {
    const float* intr     = (const float*)d_in[0];  // (B,3,3)
    const float* feat     = (const float*)d_in[1];  // (B,C,H,W)
    const float* intr_inv = (const float*)d_in[2];  // (B,3,3)
    const float* srcp     = (const float*)d_in[3];  // (B,4,4)
    const float* refp     = (const float*)d_in[4];  // (B,4,4)
    const float* depth    = (const float*)d_in[5];  // (B,D,H,W)
    float* out = (float*)d_out;
    float* ws  = (float*)d_ws;                      // 24 floats used

    setup_rt_kernel<<<1, 32, 0, stream>>>(intr, intr_inv, srcp, refp, ws);

    dim3 grid(HW / (256 * 4), BB * DD);             // (20, 64)
    warp_bilinear_kernel<<<grid, 256, 0, stream>>>(feat, depth, ws, out);
}